// HiPool_6408091206268
// MI455X (gfx1250) — compile-verified
//
#include <hip/hip_runtime.h>
#include <math.h>

typedef __attribute__((ext_vector_type(2))) float v2f;
typedef __attribute__((ext_vector_type(8))) float v8f;

#define NN   8192
#define IND  768
#define HIDD 128
#define NC1  10
#define NC2  5
#define P1   820   // ceil(8192/10)

#define PCHUNKS 4      // row chunks for pooling sum
#define PROWS   205    // rows per pooling chunk (820/4)
#define GKCH    512    // K-chunk for the alpha@x GEMM
#define GCHUNKS 16     // 8192/512

// ---------------------------------------------------------------- init
__global__ void zero_adj(int* __restrict__ a) {
    if (threadIdx.x < 128) a[threadIdx.x] = 0;
}

// ------------------------------------------- adj1 = s1^T A s1 (block sums)
// One block per adjacency row. Each thread takes a CONTIGUOUS run of 8 int4
// (32 ints); cluster boundaries are int4-aligned (820 = 4*205) and a 32-int
// run crosses at most one boundary -> 1-2 LDS atomics per thread.
__global__ void adj_block_sum(const int* __restrict__ A, int* __restrict__ adj1) {
    __shared__ int sacc[16];
    const int tid = threadIdx.x;
    const int row = blockIdx.x;
    if (tid < 16) sacc[tid] = 0;
    __syncthreads();
    const int r = row / P1;
    const int4* p = reinterpret_cast<const int4*>(A + (size_t)row * NN);
    const int base4 = tid * 8;                // int4 index, 2048 total
    int c0 = (base4 * 4) / P1;
    int acc = 0;
#pragma unroll
    for (int u = 0; u < 8; ++u) {
        int4 v = p[base4 + u];
        int c = ((base4 + u) * 4) / P1;
        if (c != c0) { atomicAdd(&sacc[c0], acc); acc = 0; c0 = c; }
        acc += v.x + v.y + v.z + v.w;
    }
    atomicAdd(&sacc[c0], acc);
    __syncthreads();
    if (tid < NC1) atomicAdd(&adj1[r * NC1 + tid], sacc[tid]);
}

// ------------------------------------------- pooling partials: s1^T x, 4 row-chunks
__global__ void pool_part(const float* __restrict__ x, float* __restrict__ part) {
    const int col = blockIdx.x * 256 + threadIdx.x;  // < 768
    const int r = blockIdx.y;                        // < 16
    const int ch = blockIdx.z;                       // < 4
    float* dst = part + ((size_t)ch * 16 + r) * IND + col;
    if (r >= NC1) { *dst = 0.f; return; }
    const int rtot = min(P1, NN - r * P1);           // 820 or 812
    const int i0 = ch * PROWS;
    const int i1 = min(i0 + PROWS, rtot);
    const float* px = x + ((size_t)(r * P1 + i0)) * IND + col;
    float s = 0.f;
    for (int i = 0; i < i1 - i0; ++i) s += px[(size_t)i * IND];
    *dst = s;
}

// ------------------------------------------- x1sum = sum of 4 pooling partials
__global__ void pool_reduce(const float* __restrict__ part, float* __restrict__ x1sum) {
    const int idx = blockIdx.x * 256 + threadIdx.x;  // < 16*768
    float s = 0.f;
#pragma unroll
    for (int c = 0; c < PCHUNKS; ++c) s += part[(size_t)c * 16 * IND + idx];
    x1sum[idx] = s;
}

// ------------------------------------------- partial D = A[16,kc] @ B[kc,N]
// grid.x = N/16 tile, grid.y = K chunk. One wave per (tile, chunk),
// fp32 WMMA 16x16x4, partial written to Dout + chunk*16*Nld.
__global__ void wmma_gemm16(const float* __restrict__ Amat, int lda, int K, int kchunk,
                            const float* __restrict__ Bmat, int Nld,
                            float* __restrict__ Dout) {
    const int n0 = blockIdx.x * 16;
    const int kbeg = blockIdx.y * kchunk;
    const int kend = min(kbeg + kchunk, K);
    const int lane = threadIdx.x;
    const int half = lane >> 4;
    const int m = lane & 15;
    v8f c;
#pragma unroll
    for (int r = 0; r < 8; ++r) c[r] = 0.f;
    const float* arow = Amat + (size_t)m * lda;
    for (int k0 = kbeg; k0 < kend; k0 += 4) {
        float4 av = *reinterpret_cast<const float4*>(arow + k0);
        v2f a, b;
        a.x = half ? av.z : av.x;
        a.y = half ? av.w : av.y;
        const int kb = k0 + 2 * half;
        b.x = Bmat[(size_t)kb * Nld + n0 + m];
        b.y = Bmat[(size_t)(kb + 1) * Nld + n0 + m];
        c = __builtin_amdgcn_wmma_f32_16x16x4_f32(false, a, false, b,
                                                  (short)0, c, false, false);
    }
    float* dchunk = Dout + (size_t)blockIdx.y * 16 * Nld;
#pragma unroll
    for (int r = 0; r < 8; ++r)
        dchunk[(size_t)(r + 8 * half) * Nld + n0 + m] = c[r];
}

// ------------------------------------------- x1new = x1sum + sum of 16 GEMM partials
__global__ void gemm_reduce(const float* __restrict__ part, const float* __restrict__ x1sum,
                            float* __restrict__ x1new) {
    const int idx = blockIdx.x * 256 + threadIdx.x;  // < 16*768
    float s = x1sum[idx];
#pragma unroll
    for (int c = 0; c < GCHUNKS; ++c) s += part[(size_t)c * 16 * IND + idx];
    x1new[idx] = s;
}

// ------------------------------------------- scores = q @ x^T, masked by (1-s1)^T
__global__ void wmma_scores(const float* __restrict__ Q, const float* __restrict__ x,
                            float* __restrict__ S) {
    const int n0 = blockIdx.x * 16;
    const int lane = threadIdx.x;
    const int half = lane >> 4;
    const int m = lane & 15;
    v8f c;
#pragma unroll
    for (int r = 0; r < 8; ++r) c[r] = 0.f;
    const float* arow = Q + (size_t)m * IND;
    const float* brow = x + (size_t)(n0 + m) * IND;  // B[k][n] = x[n][k]
    for (int k0 = 0; k0 < IND; k0 += 4) {
        float4 av = *reinterpret_cast<const float4*>(arow + k0);
        float4 bv = *reinterpret_cast<const float4*>(brow + k0);
        v2f a, b;
        a.x = half ? av.z : av.x; a.y = half ? av.w : av.y;
        b.x = half ? bv.z : bv.x; b.y = half ? bv.w : bv.y;
        c = __builtin_amdgcn_wmma_f32_16x16x4_f32(false, a, false, b,
                                                  (short)0, c, false, false);
    }
    const int n = n0 + m;
    const int cn = n / P1;
#pragma unroll
    for (int r = 0; r < 8; ++r) {
        const int mr = r + 8 * half;
        float v = c[r];
        if (cn == mr) v = 0.f;  // multiplicative mask, as in reference
        S[(size_t)mr * NN + n] = v;
    }
}

// ------------------------------------------- row softmax over N=8192 (rows>=10 -> 0)
__global__ void softmax_rows(float* __restrict__ S) {
    const int m = blockIdx.x;
    const int tid = threadIdx.x;
    float* row = S + (size_t)m * NN;
    if (m >= NC1) {
        for (int j = tid; j < NN; j += 256) row[j] = 0.f;
        return;
    }
    __shared__ float red[256];
    float mx = -3.4e38f;
    for (int j = tid; j < NN; j += 256) mx = fmaxf(mx, row[j]);
    red[tid] = mx; __syncthreads();
    for (int o = 128; o > 0; o >>= 1) {
        if (tid < o) red[tid] = fmaxf(red[tid], red[tid + o]);
        __syncthreads();
    }
    const float rmax = red[0]; __syncthreads();
    float sm = 0.f;
    for (int j = tid; j < NN; j += 256) sm += expf(row[j] - rmax);
    red[tid] = sm; __syncthreads();
    for (int o = 128; o > 0; o >>= 1) {
        if (tid < o) red[tid] += red[tid + o];
        __syncthreads();
    }
    const float inv = 1.f / red[0];
    for (int j = tid; j < NN; j += 256) row[j] = expf(row[j] - rmax) * inv;
}

// ------------------------------------------- everything tiny, one block of 128
__global__ void tail_kernel(const float* __restrict__ x1,   // [16][768]
                            const int* __restrict__ adj1_i, // [10][10]
                            const float* __restrict__ W1, const float* __restrict__ b1,
                            const float* __restrict__ Wc2,
                            const float* __restrict__ W2, const float* __restrict__ b2,
                            const float* __restrict__ linW, const float* __restrict__ linb,
                            float* __restrict__ out8) {
    __shared__ float a1[NC1][NC1], an1[NC1][NC1], d1[NC1];
    __shared__ float t1[NC1][HIDD], h[NC1][HIDD];
    __shared__ float x2[NC2][HIDD], q2[NC2][HIDD], x2b[NC2][HIDD];
    __shared__ float t2[NC2][HIDD], om[NC2][HIDD];
    __shared__ float a2[NC2][NC2], an2[NC2][NC2], d2[NC2];
    __shared__ float sc2[NC2][NC1];
    __shared__ float pooled[HIDD];
    __shared__ float z[8];
    const int t = threadIdx.x;  // 128 threads

    if (t < 100) a1[t / 10][t % 10] = (float)adj1_i[t];
    __syncthreads();
    if (t < NC1) {
        float s = 0.f;
        for (int j = 0; j < NC1; ++j) s += (j == t) ? 1.f : a1[t][j];
        d1[t] = rsqrtf(fmaxf(s, 1.f));
    }
    __syncthreads();
    if (t < 100) {
        int i = t / 10, j = t % 10;
        float v = (i == j) ? 1.f : a1[i][j];
        an1[i][j] = d1[i] * v * d1[j];
    }
    __syncthreads();
    // t1 = x1 @ W1
    for (int i = 0; i < NC1; ++i) {
        float s = 0.f;
        for (int k = 0; k < IND; ++k) s += x1[i * IND + k] * W1[k * HIDD + t];
        t1[i][t] = s;
    }
    __syncthreads();
    // h = relu(an1 @ t1 + b1)
    for (int i = 0; i < NC1; ++i) {
        float s = b1[t];
        for (int j = 0; j < NC1; ++j) s += an1[i][j] * t1[j][t];
        h[i][t] = fmaxf(s, 0.f);
    }
    __syncthreads();
    for (int cc = 0; cc < NC2; ++cc) x2[cc][t] = h[2 * cc][t] + h[2 * cc + 1][t];
    if (t < 25) {  // adj2 uses ORIGINAL adj1 (pre-diag, pre-norm)
        int aa = t / 5, bb = t % 5;
        a2[aa][bb] = a1[2 * aa][2 * bb] + a1[2 * aa][2 * bb + 1] +
                     a1[2 * aa + 1][2 * bb] + a1[2 * aa + 1][2 * bb + 1];
    }
    __syncthreads();
    // q2 = x2 @ Wc2
    for (int m = 0; m < NC2; ++m) {
        float s = 0.f;
        for (int k = 0; k < HIDD; ++k) s += x2[m][k] * Wc2[k * HIDD + t];
        q2[m][t] = s;
    }
    __syncthreads();
    if (t < 50) {
        int m = t / 10, n = t % 10;
        float s = 0.f;
        for (int k = 0; k < HIDD; ++k) s += q2[m][k] * h[n][k];
        if (n / 2 == m) s = 0.f;  // multiplicative mask (1-s2)^T
        sc2[m][n] = s;
    }
    __syncthreads();
    if (t < NC2) {  // softmax over 10
        float mx = sc2[t][0];
        for (int n = 1; n < NC1; ++n) mx = fmaxf(mx, sc2[t][n]);
        float sm = 0.f;
        for (int n = 0; n < NC1; ++n) sm += expf(sc2[t][n] - mx);
        float inv = 1.f / sm;
        for (int n = 0; n < NC1; ++n) sc2[t][n] = expf(sc2[t][n] - mx) * inv;
    }
    __syncthreads();
    for (int m = 0; m < NC2; ++m) {
        float s = x2[m][t];
        for (int n = 0; n < NC1; ++n) s += sc2[m][n] * h[n][t];
        x2b[m][t] = s;
    }
    if (t < NC2) {
        float s = 0.f;
        for (int j = 0; j < NC2; ++j) s += (j == t) ? 1.f : a2[t][j];
        d2[t] = rsqrtf(fmaxf(s, 1.f));
    }
    __syncthreads();
    if (t < 25) {
        int i = t / 5, j = t % 5;
        float v = (i == j) ? 1.f : a2[i][j];
        an2[i][j] = d2[i] * v * d2[j];
    }
    __syncthreads();
    for (int m = 0; m < NC2; ++m) {
        float s = 0.f;
        for (int k = 0; k < HIDD; ++k) s += x2b[m][k] * W2[k * HIDD + t];
        t2[m][t] = s;
    }
    __syncthreads();
    for (int i = 0; i < NC2; ++i) {
        float s = b2[t];
        for (int j = 0; j < NC2; ++j) s += an2[i][j] * t2[j][t];
        om[i][t] = s;  // no relu on final conv
    }
    __syncthreads();
    {
        float s = 0.f;
        for (int i = 0; i < NC2; ++i) s += om[i][t];
        pooled[t] = s * 0.2f;
    }
    __syncthreads();
    if (t < 8) {
        float s = linb[t];
        for (int k = 0; k < HIDD; ++k) s += pooled[k] * linW[k * 8 + t];
        z[t] = fmaxf(s, 0.f);
    }
    __syncthreads();
    if (t == 0) {
        float mx = z[0];
        for (int o = 1; o < 8; ++o) mx = fmaxf(mx, z[o]);
        float sm = 0.f;
        for (int o = 0; o < 8; ++o) sm += expf(z[o] - mx);
        float ls = logf(sm);
        for (int o = 0; o < 8; ++o) out8[o] = z[o] - mx - ls;
    }
}

extern "C" void kernel_launch(void* const* d_in, const int* in_sizes, int n_in,
                              void* d_out, int out_size, void* d_ws, size_t ws_size,
                              hipStream_t stream) {
    const float* x    = (const float*)d_in[0];
    const int*   ei   = (const int*)d_in[1];
    const float* Wc1  = (const float*)d_in[2];
    const float* Wc2  = (const float*)d_in[3];
    const float* W1   = (const float*)d_in[4];
    const float* b1   = (const float*)d_in[5];
    const float* W2   = (const float*)d_in[6];
    const float* b2   = (const float*)d_in[7];
    const float* linW = (const float*)d_in[8];
    const float* linb = (const float*)d_in[9];
    float* out = (float*)d_out;
    const int* A = ei + (size_t)NN * NN;  // edge_index[1]

    char* ws = (char*)d_ws;
    int*   adj1_i   = (int*)(ws);                       // 512 B
    float* x1sum    = (float*)(ws + 512);               // 16*768 f (48 KB)
    float* q        = (float*)(ws + 512 + 1 * 49152);   // 16*768 f
    float* x1new    = (float*)(ws + 512 + 2 * 49152);   // 16*768 f
    float* S        = (float*)(ws + 512 + 3 * 49152);   // 16*8192 f (512 KB)
    float* poolPart = (float*)(ws + 672256);            // 4*16*768 f (192 KB)
    float* gemmPart = (float*)(ws + 868864);            // 16*16*768 f (768 KB)

    hipLaunchKernelGGL(zero_adj, dim3(1), dim3(128), 0, stream, adj1_i);
    hipLaunchKernelGGL(adj_block_sum, dim3(NN), dim3(256), 0, stream, A, adj1_i);
    hipLaunchKernelGGL(pool_part, dim3(3, 16, PCHUNKS), dim3(256), 0, stream, x, poolPart);
    hipLaunchKernelGGL(pool_reduce, dim3(48), dim3(256), 0, stream, poolPart, x1sum);
    // q = x1sum @ Wc1   (single K chunk)
    hipLaunchKernelGGL(wmma_gemm16, dim3(IND / 16, 1), dim3(32), 0, stream,
                       x1sum, IND, IND, IND, Wc1, IND, q);
    hipLaunchKernelGGL(wmma_scores, dim3(NN / 16), dim3(32), 0, stream, q, x, S);
    hipLaunchKernelGGL(softmax_rows, dim3(16), dim3(256), 0, stream, S);
    // partials of alpha @ x   (16 K chunks of 512)
    hipLaunchKernelGGL(wmma_gemm16, dim3(IND / 16, GCHUNKS), dim3(32), 0, stream,
                       S, NN, NN, GKCH, x, IND, gemmPart);
    hipLaunchKernelGGL(gemm_reduce, dim3(48), dim3(256), 0, stream, gemmPart, x1sum, x1new);
    hipLaunchKernelGGL(tail_kernel, dim3(1), dim3(128), 0, stream,
                       x1new, adj1_i, W1, b1, Wc2, W2, b2, linW, linb, out);
}